// SparseVectorQuantizer_75539884802812
// MI455X (gfx1250) — compile-verified
//
#include <hip/hip_runtime.h>

typedef float v2f __attribute__((ext_vector_type(2)));
typedef float v8f __attribute__((ext_vector_type(8)));

#define N_VOX 65536
#define K_CB  8192
#define D_EMB 64

#define WAVES_PER_BLOCK 8
#define ROWS_PER_WAVE   32              // two 16-row WMMA M-tiles per wave
#define NUM_BLOCKS (N_VOX / (WAVES_PER_BLOCK * ROWS_PER_WAVE))   // 256
#define NUM_WAVES  (NUM_BLOCKS * WAVES_PER_BLOCK)                // 2048

// ws layout (floats): [0, 8192) = e_sq ; [8192, 8192+NUM_WAVES) = loss partials
#define WS_ESQ   0
#define WS_PART  8192

// ---------------------------------------------------------------------------
// Kernel 1: e_sq[k] = sum_d codebook[k][d]^2
// ---------------------------------------------------------------------------
__global__ __launch_bounds__(256) void vq_esq_kernel(const float* __restrict__ cb,
                                                     float* __restrict__ esq) {
  int k = blockIdx.x * blockDim.x + threadIdx.x;
  if (k < K_CB) {
    const float4* row = reinterpret_cast<const float4*>(cb + (size_t)k * D_EMB);
    float s = 0.f;
#pragma unroll
    for (int j = 0; j < D_EMB / 4; ++j) {
      float4 v = row[j];
      s += v.x * v.x + v.y * v.y + v.z * v.z + v.w * v.w;
    }
    esq[k] = s;
  }
}

// ---------------------------------------------------------------------------
// Kernel 2: WMMA distance GEMM + argmin + gather + quantized + loss partials.
// Each wave owns 32 z-rows (two A panels). Per codebook tile the full B panel
// is preloaded into registers (16 claused b64 loads), then two independent
// WMMA chains consume it (2 WMMA per load, progressive loadcnt waits).
// ---------------------------------------------------------------------------
__global__ __launch_bounds__(256) void vq_main_kernel(const float* __restrict__ z,
                                                      const float* __restrict__ cb,
                                                      const float* __restrict__ esq_g,
                                                      float* __restrict__ out,
                                                      float* __restrict__ partials) {
  __shared__ float esq[K_CB];                              // 32 KB
  __shared__ int   idx_lds[WAVES_PER_BLOCK * ROWS_PER_WAVE];

  const int tid = threadIdx.x;
  for (int i = tid; i < K_CB; i += 256) esq[i] = esq_g[i];
  __syncthreads();

  const int wv   = tid >> 5;   // wave in block (0..7)
  const int lane = tid & 31;
  const int n    = lane & 15;  // A-row / B-column within 16-wide tile
  const int h    = lane >> 4;  // K-pair select for 16x16x4 fragments

  const int m0 = (blockIdx.x * WAVES_PER_BLOCK + wv) * ROWS_PER_WAVE;

  // --- Preload two A panels: rows m0..m0+15 and m0+16..m0+31.
  // A(16x4 f32) frag chunk j: lane(n,h) holds z[row][4j+2h], z[row][4j+2h+1].
  const v2f* zrow0 = reinterpret_cast<const v2f*>(z + (size_t)(m0 + n) * D_EMB);
  const v2f* zrow1 = reinterpret_cast<const v2f*>(z + (size_t)(m0 + 16 + n) * D_EMB);
  v2f a0[16], a1[16];
#pragma unroll
  for (int j = 0; j < 16; ++j) { a0[j] = zrow0[2 * j + h]; a1[j] = zrow1[2 * j + h]; }

  float bestS[16];
  int   bestI[16];
#pragma unroll
  for (int r = 0; r < 16; ++r) { bestS[r] = 3.4e38f; bestI[r] = 0; }

  const v2f* cb2 = reinterpret_cast<const v2f*>(cb);

  // --- Sweep all 512 codebook N-tiles.
  for (int k0 = 0; k0 < K_CB; k0 += 16) {
    const v2f* brow = cb2 + (size_t)(k0 + n) * (D_EMB / 2);

    // Stage the whole B panel in registers: 16 b64 loads, claused together.
    v2f b[16];
#pragma unroll
    for (int j = 0; j < 16; ++j) b[j] = brow[2 * j + h];
    // Keep all loads issued before any WMMA consumes them (prevents the
    // scheduler from collapsing into load->wait0->wmma serialization).
    __builtin_amdgcn_sched_barrier(0);

    v8f c0 = {}, c1 = {};
#pragma unroll
    for (int j = 0; j < 16; ++j) {
      // Two independent accumulation chains sharing one B fragment.
      c0 = __builtin_amdgcn_wmma_f32_16x16x4_f32(false, a0[j], false, b[j],
                                                 (short)0, c0, false, false);
      c1 = __builtin_amdgcn_wmma_f32_16x16x4_f32(false, a1[j], false, b[j],
                                                 (short)0, c1, false, false);
    }
    __builtin_amdgcn_sched_barrier(0);

    const float ecol = esq[k0 + n];
    const int   kidx = k0 + n;
#pragma unroll
    for (int r = 0; r < 8; ++r) {
      // squared distance minus constant ||z||^2: e_sq - 2*dot
      float s0 = fmaf(-2.0f, c0[r], ecol);
      float s1 = fmaf(-2.0f, c1[r], ecol);
      // k strictly increases -> strict '<' keeps first occurrence per lane
      if (s0 < bestS[r])     { bestS[r]     = s0; bestI[r]     = kidx; }
      if (s1 < bestS[8 + r]) { bestS[8 + r] = s1; bestI[8 + r] = kidx; }
    }
  }

  // --- Argmin butterfly within each 16-lane half (tie-break: lowest index).
#pragma unroll
  for (int off = 8; off >= 1; off >>= 1) {
#pragma unroll
    for (int r = 0; r < 16; ++r) {
      float os = __shfl_xor(bestS[r], off, 32);
      int   oi = __shfl_xor(bestI[r], off, 32);
      if (os < bestS[r] || (os == bestS[r] && oi < bestI[r])) {
        bestS[r] = os; bestI[r] = oi;
      }
    }
  }
  // C layout: lanes 0-15 hold rows {r, 16+r}; lanes 16-31 hold rows {8+r, 24+r}.
  if (lane == 0) {
#pragma unroll
    for (int r = 0; r < 8; ++r) {
      idx_lds[wv * ROWS_PER_WAVE + r]      = bestI[r];
      idx_lds[wv * ROWS_PER_WAVE + 16 + r] = bestI[8 + r];
    }
  }
  if (lane == 16) {
#pragma unroll
    for (int r = 0; r < 8; ++r) {
      idx_lds[wv * ROWS_PER_WAVE + 8 + r]  = bestI[r];
      idx_lds[wv * ROWS_PER_WAVE + 24 + r] = bestI[8 + r];
    }
  }
  __syncthreads();

  // --- Gather quantized rows, write outputs, accumulate loss partial.
  float* outv   = out + (size_t)N_VOX * D_EMB; // [vq_loss, commitment_loss]
  float* outidx = outv + 2;                    // indices_st [N]
  const v2f* z2 = reinterpret_cast<const v2f*>(z);
  v2f* outq2    = reinterpret_cast<v2f*>(out);

  float lsum = 0.f;
#pragma unroll 4
  for (int m = 0; m < ROWS_PER_WAVE; ++m) {
    const int idx = idx_lds[wv * ROWS_PER_WAVE + m];
    const int row = m0 + m;
    v2f q  = cb2[(size_t)idx * (D_EMB / 2) + lane]; // 2 cols per lane
    v2f zv = z2[(size_t)row * (D_EMB / 2) + lane];
    outq2[(size_t)row * (D_EMB / 2) + lane] = q;    // quantized = codebook[idx]
    float dx = zv.x - q.x, dy = zv.y - q.y;
    lsum += dx * dx + dy * dy;
    if (lane == 0) outidx[row] = (float)idx;
  }
#pragma unroll
  for (int off = 16; off >= 1; off >>= 1) lsum += __shfl_xor(lsum, off, 32);
  if (lane == 0) partials[blockIdx.x * WAVES_PER_BLOCK + wv] = lsum;
}

// ---------------------------------------------------------------------------
// Kernel 3: deterministic fixed-order reduction of loss partials.
// ---------------------------------------------------------------------------
__global__ __launch_bounds__(256) void vq_loss_kernel(const float* __restrict__ partials,
                                                      float* __restrict__ out) {
  __shared__ float acc[256];
  const int tid = threadIdx.x;
  float s = 0.f;
  for (int i = tid; i < NUM_WAVES; i += 256) s += partials[i];
  acc[tid] = s;
  __syncthreads();
  for (int off = 128; off >= 1; off >>= 1) {
    if (tid < off) acc[tid] += acc[tid + off];
    __syncthreads();
  }
  if (tid == 0) {
    float mean = acc[0] / (float)((size_t)N_VOX * D_EMB);
    out[(size_t)N_VOX * D_EMB]     = mean; // vq_loss (forward)
    out[(size_t)N_VOX * D_EMB + 1] = mean; // commitment_loss (forward, identical)
  }
}

// ---------------------------------------------------------------------------
extern "C" void kernel_launch(void* const* d_in, const int* in_sizes, int n_in,
                              void* d_out, int out_size, void* d_ws, size_t ws_size,
                              hipStream_t stream) {
  const float* z  = (const float*)d_in[0]; // [N, D]
  const float* cb = (const float*)d_in[1]; // [K, D]
  float* out      = (float*)d_out;
  float* ws       = (float*)d_ws;
  float* esq      = ws + WS_ESQ;
  float* partials = ws + WS_PART;

  vq_esq_kernel<<<K_CB / 256, 256, 0, stream>>>(cb, esq);
  vq_main_kernel<<<NUM_BLOCKS, 256, 0, stream>>>(z, cb, esq, out, partials);
  vq_loss_kernel<<<1, 256, 0, stream>>>(partials, out);
}